// HeteroLinear_82351702933578
// MI455X (gfx1250) — compile-verified
//
#include <hip/hip_runtime.h>

// HeteroLinear: out[b,n,o] = sum_i x[b,n,i] * W[type[n],o,i] + bias[type[n],o]
// B=16 (== WMMA M), N=16384, F_IN=F_OUT=128, 16 types.
//
// Pass 1: bin nodes by type (counts -> prefix -> scatter) into d_ws.
// Pass 2: one workgroup per (type, 64-node chunk): DMA W[t] into LDS once via
//         the Tensor Data Mover (padded rows -> bank-conflict-free), then 8
//         waves each run one node's C[16x128] = A[16x128] x W[t]^T + bias as
//         8 tiles x 32 K-steps of V_WMMA_F32_16X16X4_F32 with B from LDS.

typedef __attribute__((ext_vector_type(2))) float v2f;
typedef __attribute__((ext_vector_type(4))) float v4f;
typedef __attribute__((ext_vector_type(8))) float v8f;
typedef __attribute__((ext_vector_type(4))) unsigned int u32x4;
typedef __attribute__((ext_vector_type(4))) int i32x4;
typedef __attribute__((ext_vector_type(8))) int i32x8;

#define NNODES   16384
#define FIN      128
#define FOUT     128
#define NTYPES   16
#define LDS_STRIDE 132                 // 128 + 4 pad floats -> all 64 banks hit once
#define NODES_PER_WG 64
#define WAVES    8
#define NODES_PER_WAVE (NODES_PER_WG / WAVES)
#define MAX_BLK_PER_TYPE (NNODES / NODES_PER_WG)   // 256
#define SMEM_BYTES ((FOUT * LDS_STRIDE + FOUT) * 4)  // W panel + bias = 68096 B

#if __has_builtin(__builtin_amdgcn_tensor_load_to_lds)
#define USE_TDM 1
#endif

// ---------------- binning pre-pass ----------------

__global__ void zero_ws(int* __restrict__ ws) {
    if (threadIdx.x < 2 * NTYPES) ws[threadIdx.x] = 0;   // counts + cursors
}

__global__ void count_types(const int* __restrict__ tv, int* __restrict__ counts) {
    const int n = blockIdx.x * blockDim.x + threadIdx.x;
    if (n < NNODES) atomicAdd(&counts[tv[n] & (NTYPES - 1)], 1);
}

__global__ void make_offsets(const int* __restrict__ counts, int* __restrict__ offsets) {
    int acc = 0;
    for (int t = 0; t < NTYPES; ++t) { offsets[t] = acc; acc += counts[t]; }
}

__global__ void scatter_nodes(const int* __restrict__ tv, int* __restrict__ cursors,
                              const int* __restrict__ offsets, int* __restrict__ list) {
    const int n = blockIdx.x * blockDim.x + threadIdx.x;
    if (n < NNODES) {
        const int t = tv[n] & (NTYPES - 1);
        const int pos = atomicAdd(&cursors[t], 1);
        list[offsets[t] + pos] = n;
    }
}

// ---------------- WMMA GEMM pass ----------------

__global__ __launch_bounds__(WAVES * 32)
void hetero_gemm(const float* __restrict__ x,       // [16, N, 128]
                 const float* __restrict__ W,       // [16, 128, 128] (out,in)
                 const float* __restrict__ bias,    // [16, 128]
                 const int*   __restrict__ counts,
                 const int*   __restrict__ offsets,
                 const int*   __restrict__ list,
                 float*       __restrict__ out)     // [16, N, 128]
{
    extern __shared__ float smem[];                 // [128*132] W panel + [128] bias

    const int t   = blockIdx.x >> 8;                // 256 chunks per type
    const int blk = blockIdx.x & (MAX_BLK_PER_TYPE - 1);
    const int cnt = counts[t];
    if (blk * NODES_PER_WG >= cnt) return;          // block-uniform exit

    const float* Wt = W + (size_t)t * FOUT * FIN;
    const int tid = threadIdx.x;

#if defined(USE_TDM)
    // ---- Tensor Data Mover: DMA the 128x128 f32 W[t] panel into LDS.
    // pad_enable with pad_interval=6 (every 128 DWORDs == one row) and
    // pad_amount=3 (4 DWORDs) reproduces the 132-float padded row stride.
    if (tid < 32) {                                 // wave 0 owns the DMA
        const unsigned long long ga = (unsigned long long)(uintptr_t)Wt;
        const unsigned lds_off = (unsigned)(uintptr_t)smem;   // LDS byte offset

        u32x4 g0 = { 1u,                            // count=1 (valid user D#)
                     lds_off,                       // lds_addr
                     (unsigned)ga,                  // global_addr[31:0]
                     (unsigned)(ga >> 32) | (2u << 30) };     // addr[56:32] | type=2

        i32x8 g1 = { (int)((2u << 16)               // data_size = 4 bytes
                         | (1u << 20)               // pad_enable
                         | (6u << 22)               // pad_interval: 128 DWORDs
                         | (3u << 25)),             // pad_amount: 4 DWORDs
                     (int)(128u << 16),             // tensor_dim0 = 128 (lo16 in hi half)
                     (int)(128u << 16),             // tensor_dim1 = 128
                     (int)(128u << 16),             // tile_dim0 = 128
                     128,                           // tile_dim1 = 128 (tile_dim2 = 0)
                     128,                           // tensor_dim0_stride = 128
                     0, 0 };                        // dim1 stride unused (2-D)

        i32x4 gz4 = { 0, 0, 0, 0 };                 // 2-D tensor: groups 2/3 inert
#if __clang_major__ >= 23
        i32x8 gz8 = { 0, 0, 0, 0, 0, 0, 0, 0 };
        __builtin_amdgcn_tensor_load_to_lds(g0, g1, gz4, gz4, gz8, 0);
#else
        __builtin_amdgcn_tensor_load_to_lds(g0, g1, gz4, gz4, 0);
#endif
        __builtin_amdgcn_s_wait_tensorcnt((short)0);
    }
#else
    // ---- Fallback: cooperative staging (coalesced b128 global -> b128 ds).
#pragma unroll
    for (int it = 0; it < 16; ++it) {
        const int c  = tid + it * 256;              // float4 chunk id, 4096 total
        const int o  = c >> 5;                      // 32 float4 per row
        const int kc = c & 31;
        v4f v = *(const v4f*)(Wt + (size_t)o * FIN + kc * 4);
        *(v4f*)(smem + o * LDS_STRIDE + kc * 4) = v;
    }
#endif

    float* sbias = smem + FOUT * LDS_STRIDE;
    if (tid < FOUT) sbias[tid] = bias[(size_t)t * FOUT + tid];
    __syncthreads();

    const int lane = tid & 31;
    const int wave = tid >> 5;
    const int m = lane & 15;                        // A row / B column base
    const int h = lane >> 4;                        // K-pair select {2h, 2h+1}
    const int seg = offsets[t];

    for (int i = 0; i < NODES_PER_WAVE; ++i) {
        const int idx = blk * NODES_PER_WG + wave * NODES_PER_WAVE + i;
        if (idx >= cnt) break;                      // wave-uniform
        const int node = list[seg + idx];

        // A strip: x[m, node, :]; fragment s supplies K = 4s + 2h + {0,1}.
        const float* xrow = x + ((size_t)m * NNODES + (size_t)node) * FIN;
        v2f a[32];
#pragma unroll
        for (int s = 0; s < 32; ++s)
            a[s] = *(const v2f*)(xrow + 4 * s + 2 * h);

        for (int ot = 0; ot < 8; ++ot) {
            const int o = ot * 16 + m;              // this lane's output column
            const float* brow = smem + o * LDS_STRIDE + 2 * h;  // conflict-free ds_load_b64

            const float bv = sbias[o];              // bias seeds the accumulator
            v8f c;
#pragma unroll
            for (int j = 0; j < 8; ++j) c[j] = bv;

#pragma unroll
            for (int s = 0; s < 32; ++s) {
                v2f bf = *(const v2f*)(brow + 4 * s);
                c = __builtin_amdgcn_wmma_f32_16x16x4_f32(
                        false, a[s], false, bf, (short)0, c, false, false);
            }

            // C/D layout: VGPR j holds row (j + 8h), column (lane & 15).
#pragma unroll
            for (int j = 0; j < 8; ++j)
                out[((size_t)(j + 8 * h) * NNODES + (size_t)node) * FOUT + o] = c[j];
        }
    }
}

// ---------------- launch ----------------

extern "C" void kernel_launch(void* const* d_in, const int* in_sizes, int n_in,
                              void* d_out, int out_size, void* d_ws, size_t ws_size,
                              hipStream_t stream) {
    const float* x        = (const float*)d_in[0];  // [16, 16384, 128]
    const int*   type_vec = (const int*)  d_in[1];  // [16384]
    const float* W        = (const float*)d_in[2];  // [16, 128, 128]
    const float* bias     = (const float*)d_in[3];  // [16, 128]
    float*       out      = (float*)d_out;          // [16, 16384, 128]

    int* ws       = (int*)d_ws;                     // needs ~66 KB of d_ws
    int* counts   = ws;
    int* cursors  = ws + 16;
    int* offsets  = ws + 32;
    int* list     = ws + 48;                        // [16384]

    zero_ws     <<<1, 64, 0, stream>>>(ws);
    count_types <<<NNODES / 256, 256, 0, stream>>>(type_vec, counts);
    make_offsets<<<1, 1, 0, stream>>>(counts, offsets);
    scatter_nodes<<<NNODES / 256, 256, 0, stream>>>(type_vec, cursors, offsets, list);

    hetero_gemm<<<NTYPES * MAX_BLK_PER_TYPE, WAVES * 32, SMEM_BYTES, stream>>>(
        x, W, bias, counts, offsets, list, out);
}